// SwitchGate_20323785244714
// MI455X (gfx1250) — compile-verified
//
#include <hip/hip_runtime.h>
#include <hip/hip_bf16.h>

typedef float v2f __attribute__((ext_vector_type(2)));
typedef float v8f __attribute__((ext_vector_type(8)));

#define NTOK      262144
#define DIMK      1024
#define NEXP      64
#define TILE_M    128      // tokens per block
#define KC        32       // k-chunk staged in LDS
#define NCHUNK    (DIMK / KC)
#define XS_STRIDE 36       // floats; 144B rows: 16B-aligned, conflict-free frag reads
#define WS_STRIDE 36
#define LG_STRIDE 67       // odd stride -> conflict-free per-token column scans
#define XBUF      (TILE_M * XS_STRIDE)   // floats per x stage buffer
#define WBUF      (NEXP   * WS_STRIDE)   // floats per W stage buffer
#define CAPACITY  262144.0f
#define EPS_V     1e-6f

// CDNA5 async global->LDS copy (ASYNCcnt path, no VGPR round-trip)
__device__ __forceinline__ void async_copy_b128(unsigned lds_off, const float* gsrc) {
    asm volatile("global_load_async_to_lds_b128 %0, %1, off"
                 :: "v"(lds_off), "v"(gsrc) : "memory");
}
__device__ __forceinline__ void wait_async0() {
    asm volatile("s_wait_asynccnt 0" ::: "memory");
}

__global__ void zero_denom_kernel(float* __restrict__ denom) {
    denom[threadIdx.x] = 0.0f;
}

__global__ __launch_bounds__(256)
void gate_pass1_kernel(const float* __restrict__ x, const float* __restrict__ W,
                       const float* __restrict__ b, float* __restrict__ out,
                       float* __restrict__ denom_g) {
    __shared__ union {
        struct { float xs[2][XBUF]; float ws[2][WBUF]; } ld;   // double-buffered staging
        float logits[TILE_M * LG_STRIDE];                      // aliases staging (post K-loop)
    } u;
    __shared__ float bias_s[NEXP];
    __shared__ float denom_s[NEXP];
    __shared__ float st_m1[TILE_M], st_m2[TILE_M], st_is[TILE_M];

    const int tid  = threadIdx.x;
    const int lane = tid & 31;
    const int wave = tid >> 5;          // 8 waves
    const int lm   = lane & 15;
    const int lh2  = (lane >> 4) * 2;   // 0 or 2
    const int wtok = wave * 16;         // this wave's token base in tile
    const int tok_base = blockIdx.x * TILE_M;

    if (tid < NEXP) { bias_s[tid] = b[tid]; denom_s[tid] = 0.0f; }

    // per-thread staging assignments (fixed across K-chunks; only src advances)
    const float* xsrc[4];
    unsigned     xdst[4];
    #pragma unroll
    for (int j = 0; j < 4; ++j) {
        int f  = tid + 256 * j;         // 0..1023 float4 slots of 128x32 tile
        int t  = f >> 3;
        int kq = f & 7;
        xsrc[j] = x + (size_t)(tok_base + t) * DIMK + kq * 4;
        xdst[j] = (unsigned)(uintptr_t)&u.ld.xs[0][t * XS_STRIDE + kq * 4];
    }
    const float* wsrc[2];
    unsigned     wdst[2];
    #pragma unroll
    for (int j = 0; j < 2; ++j) {
        int f  = tid + 256 * j;         // 0..511 float4 slots of 64x32 tile
        int e  = f >> 3;
        int kq = f & 7;
        wsrc[j] = W + (size_t)e * DIMK + kq * 4;
        wdst[j] = (unsigned)(uintptr_t)&u.ld.ws[0][e * WS_STRIDE + kq * 4];
    }

    // per-wave LDS fragment row pointers (buffer 0 base)
    const float* xrow  = &u.ld.xs[0][(wtok + lm) * XS_STRIDE];
    const float* wrow0 = &u.ld.ws[0][( 0 + lm) * WS_STRIDE];
    const float* wrow1 = &u.ld.ws[0][(16 + lm) * WS_STRIDE];
    const float* wrow2 = &u.ld.ws[0][(32 + lm) * WS_STRIDE];
    const float* wrow3 = &u.ld.ws[0][(48 + lm) * WS_STRIDE];

    v8f acc[4] = {};                    // 16 tokens x 64 experts per wave

    // prologue: prefetch chunk 0 into buffer 0
    #pragma unroll
    for (int j = 0; j < 4; ++j) async_copy_b128(xdst[j], xsrc[j]);
    #pragma unroll
    for (int j = 0; j < 2; ++j) async_copy_b128(wdst[j], wsrc[j]);
    #pragma unroll
    for (int j = 0; j < 4; ++j) xsrc[j] += KC;
    #pragma unroll
    for (int j = 0; j < 2; ++j) wsrc[j] += KC;

    for (int ic = 0; ic < NCHUNK; ++ic) {
        const int cur = ic & 1;
        const int nxt = cur ^ 1;

        wait_async0();                  // chunk ic landed (this wave's copies)
        __syncthreads();                // ... and everyone else's

        if (ic + 1 < NCHUNK) {          // prefetch chunk ic+1 into the other buffer
            #pragma unroll
            for (int j = 0; j < 4; ++j) async_copy_b128(xdst[j] + nxt * (XBUF * 4), xsrc[j]);
            #pragma unroll
            for (int j = 0; j < 2; ++j) async_copy_b128(wdst[j] + nxt * (WBUF * 4), wsrc[j]);
            #pragma unroll
            for (int j = 0; j < 4; ++j) xsrc[j] += KC;
            #pragma unroll
            for (int j = 0; j < 2; ++j) wsrc[j] += KC;
        }

        const int xo = cur * XBUF;      // float offsets into current buffers
        const int wo = cur * WBUF;

        #pragma unroll
        for (int ks = 0; ks < KC; ks += 4) {
            int kk = ks + lh2;
            // batch all fragment loads before the WMMA group
            v2f a  = *(const v2f*)(xrow  + xo + kk);  // A 16x4: lanes 0-15 K={0,1}, 16-31 K={2,3}
            v2f b0 = *(const v2f*)(wrow0 + wo + kk);  // B 4x16: B[k][n] = W[n][k]
            v2f b1 = *(const v2f*)(wrow1 + wo + kk);
            v2f b2 = *(const v2f*)(wrow2 + wo + kk);
            v2f b3 = *(const v2f*)(wrow3 + wo + kk);
            acc[0] = __builtin_amdgcn_wmma_f32_16x16x4_f32(false, a, false, b0, (short)0, acc[0], false, false);
            acc[1] = __builtin_amdgcn_wmma_f32_16x16x4_f32(false, a, false, b1, (short)0, acc[1], false, false);
            acc[2] = __builtin_amdgcn_wmma_f32_16x16x4_f32(false, a, false, b2, (short)0, acc[2], false, false);
            acc[3] = __builtin_amdgcn_wmma_f32_16x16x4_f32(false, a, false, b3, (short)0, acc[3], false, false);
        }
        __syncthreads();                // readers done before this buffer is refilled
    }

    // spill logits (+bias) to LDS; C layout: VGPR r -> M = r + 8*(lane>>4), lane%16 -> N
    #pragma unroll
    for (int t = 0; t < 4; ++t) {
        float bv = bias_s[t * 16 + lm];
        #pragma unroll
        for (int r = 0; r < 8; ++r) {
            int tl = wtok + (lh2 >> 1) * 8 + r;
            u.logits[tl * LG_STRIDE + t * 16 + lm] = acc[t][r] + bv;
        }
    }
    __syncthreads();

    // per-token softmax stats + top-2 threshold (softmax is monotone in logits)
    if (tid < TILE_M) {
        const float* lrow = &u.logits[tid * LG_STRIDE];
        float m1 = -1e30f, m2 = -1e30f;
        for (int e = 0; e < NEXP; ++e) {
            float v = lrow[e];
            if (v > m1)      { m2 = m1; m1 = v; }
            else if (v > m2) { m2 = v; }
        }
        float s = 0.0f;
        for (int e = 0; e < NEXP; ++e) s += __expf(lrow[e] - m1);
        st_m1[tid] = m1; st_m2[tid] = m2; st_is[tid] = 1.0f / s;
    }
    __syncthreads();

    // cooperative masked write (coalesced float4) + per-expert partial sums
    #pragma unroll
    for (int j = 0; j < 8; ++j) {
        int q  = tid + 256 * j;         // 0..2047 -> (token, 4-expert quad)
        int t  = q >> 4;
        int eq = (q & 15) * 4;
        float m1 = st_m1[t], m2 = st_m2[t], is = st_is[t];
        float vv[4];
        #pragma unroll
        for (int c = 0; c < 4; ++c) {
            float v    = u.logits[t * LG_STRIDE + eq + c];
            float mval = (v >= m2) ? __expf(v - m1) * is : 0.0f;
            vv[c] = mval;
            if (mval != 0.0f) atomicAdd(&denom_s[eq + c], mval);   // ds_add_f32
        }
        float4 o = {vv[0], vv[1], vv[2], vv[3]};
        *(float4*)(out + (size_t)(tok_base + t) * NEXP + eq) = o;
    }
    __syncthreads();
    if (tid < NEXP) atomicAdd(&denom_g[tid], denom_s[tid]);
}

__global__ void finalize_kernel(float* __restrict__ ws) {
    int e = threadIdx.x;
    ws[NEXP + e] = CAPACITY / (ws[e] + EPS_V);
}

__global__ __launch_bounds__(256)
void scale_kernel(float4* __restrict__ out4, const float* __restrict__ scale) {
    __shared__ float s_s[NEXP];
    if (threadIdx.x < NEXP) s_s[threadIdx.x] = scale[threadIdx.x];
    __syncthreads();
    size_t idx = (size_t)blockIdx.x * 256 + threadIdx.x;
    int eq = (int)(idx & 15) * 4;       // 16 float4 per 64-expert row
    float4 v = out4[idx];
    v.x *= s_s[eq + 0];
    v.y *= s_s[eq + 1];
    v.z *= s_s[eq + 2];
    v.w *= s_s[eq + 3];
    out4[idx] = v;
}

extern "C" void kernel_launch(void* const* d_in, const int* in_sizes, int n_in,
                              void* d_out, int out_size, void* d_ws, size_t ws_size,
                              hipStream_t stream) {
    const float* x = (const float*)d_in[0];
    const float* W = (const float*)d_in[1];
    const float* b = (const float*)d_in[2];
    float* out = (float*)d_out;
    float* ws  = (float*)d_ws;          // [0:64) denom, [64:128) scale

    zero_denom_kernel<<<1, NEXP, 0, stream>>>(ws);
    gate_pass1_kernel<<<NTOK / TILE_M, 256, 0, stream>>>(x, W, b, out, ws);
    finalize_kernel<<<1, NEXP, 0, stream>>>(ws);
    scale_kernel<<<(NTOK / 4) * (NEXP / 256), 256, 0, stream>>>((float4*)out, ws + NEXP);
}